// PolicyGradientLossEnrollment_28260884807716
// MI455X (gfx1250) — compile-verified
//
#include <hip/hip_runtime.h>
#include <hip/hip_bf16.h>
#include <stdint.h>

#define B_ROWS 4096
#define M_COLS 512
#define NUM_MC 25
#define K_TOP 32
#define NTASK (B_ROWS * NUM_MC)                 /* 102400 */
#define GUM_N2 ((uint64_t)NTASK * M_COLS / 2)   /* 26214400 */
#define PERM_N2 ((uint64_t)NTASK * K_TOP / 2)   /* 1638400 */

typedef __attribute__((ext_vector_type(2))) float v2f;
typedef __attribute__((ext_vector_type(8))) float v8f;

struct U2 { uint32_t a, b; };

static __host__ __device__ inline uint32_t rotl32(uint32_t x, int r) {
  return (x << r) | (x >> (32 - r));
}

// Threefry-2x32, 20 rounds, JAX-compatible rotation/key schedule.
static __host__ __device__ inline U2 threefry2x32(uint32_t k0, uint32_t k1,
                                                  uint32_t x0, uint32_t x1) {
  const uint32_t ks2 = 0x1BD11BDAu ^ k0 ^ k1;
  x0 += k0; x1 += k1;
#define TF_R(r) { x0 += x1; x1 = rotl32(x1, (r)); x1 ^= x0; }
  TF_R(13) TF_R(15) TF_R(26) TF_R(6)
  x0 += k1;  x1 += ks2 + 1u;
  TF_R(17) TF_R(29) TF_R(16) TF_R(24)
  x0 += ks2; x1 += k0 + 2u;
  TF_R(13) TF_R(15) TF_R(26) TF_R(6)
  x0 += k0;  x1 += k1 + 3u;
  TF_R(17) TF_R(29) TF_R(16) TF_R(24)
  x0 += k1;  x1 += ks2 + 4u;
  TF_R(13) TF_R(15) TF_R(26) TF_R(6)
  x0 += ks2; x1 += k0 + 5u;
#undef TF_R
  U2 r; r.a = x0; r.b = x1; return r;
}

// JAX random_bits flat mapping: counts split into halves (x0=i, x1=i+N/2).
static __device__ inline uint32_t jax_bits(uint32_t k0, uint32_t k1,
                                           uint64_t n, uint64_t n2) {
  uint32_t x0, x1; int w;
  if (n < n2) { x0 = (uint32_t)n;        x1 = (uint32_t)(n + n2); w = 0; }
  else        { x0 = (uint32_t)(n - n2); x1 = (uint32_t)(n);      w = 1; }
  U2 r = threefry2x32(k0, k1, x0, x1);
  return w ? r.b : r.a;
}

static __device__ inline float u01_from_bits(uint32_t bits) {
  uint32_t u = (bits >> 9) | 0x3F800000u;
  float f; __builtin_memcpy(&f, &u, 4);
  return f - 1.0f;  // [0, 1)
}

static __device__ inline float gumbel_from_bits(uint32_t bits) {
  const float TINY = 1.17549435e-38f;  // jnp.finfo(float32).tiny
  float f = u01_from_bits(bits);
  float u = f * (1.0f - TINY) + TINY;
  u = fmaxf(u, TINY);
  return -logf(-logf(u));
}

// ---------------- Kernel A: per-row logsumexp + relevance sum ---------------
__global__ __launch_bounds__(512)
void row_stats_kernel(const float* __restrict__ score,
                      const float* __restrict__ rel,
                      float* __restrict__ rowLse,
                      float* __restrict__ rowRelSum) {
  __shared__ float red[512];
  const int tid = threadIdx.x, b = blockIdx.x;
  float s = score[b * M_COLS + tid];
  red[tid] = s; __syncthreads();
  for (int st = 256; st > 0; st >>= 1) {
    if (tid < st) red[tid] = fmaxf(red[tid], red[tid + st]);
    __syncthreads();
  }
  float m = red[0]; __syncthreads();
  red[tid] = expf(s - m); __syncthreads();
  for (int st = 256; st > 0; st >>= 1) {
    if (tid < st) red[tid] += red[tid + st];
    __syncthreads();
  }
  float z = red[0]; __syncthreads();
  red[tid] = rel[b * M_COLS + tid]; __syncthreads();
  for (int st = 256; st > 0; st >>= 1) {
    if (tid < st) red[tid] += red[tid + st];
    __syncthreads();
  }
  if (tid == 0) {
    rowLse[b] = m + logf(z);
    rowRelSum[b] = red[0];
  }
}

// ------------- Kernel B: one (b, mc) Plackett-Luce sample per block ---------
__global__ __launch_bounds__(512)
void mc_task_kernel(const float* __restrict__ score,
                    const float* __restrict__ rel,
                    const float* __restrict__ rowLse,
                    const float* __restrict__ rowRelSum,
                    uint32_t kg0, uint32_t kg1, uint32_t kp0, uint32_t kp1,
                    float* __restrict__ gout) {
  __shared__ float s_key[512];
  __shared__ int   s_idx[512];
  __shared__ float s_lp[512];
  __shared__ float s_rel[512];
  __shared__ float s_rs[512];
  __shared__ float s_dn[512];
  const int tid = threadIdx.x;
  const int bid = blockIdx.x;
  const int brow = bid / NUM_MC;

  // 1) log-probs and normalized relevance (NaN -> 0, like jnp.where(isnan))
  const float lse = rowLse[brow];
  const float rsum = rowRelSum[brow];
  const float lp = score[brow * M_COLS + tid] - lse;
  s_lp[tid] = lp;
  float rv = rel[brow * M_COLS + tid] / rsum;
  if (rv != rv) rv = 0.0f;
  s_rel[tid] = rv;

  // 2) Gumbel-perturbed key
  const uint64_t n = (uint64_t)bid * M_COLS + (uint64_t)tid;
  s_key[tid] = lp + gumbel_from_bits(jax_bits(kg0, kg1, n, GUM_N2));
  s_idx[tid] = tid;
  __syncthreads();

  // 3) bitonic argsort, descending by key, index tiebreak (stable-equivalent)
  for (int k = 2; k <= 512; k <<= 1) {
    for (int j = k >> 1; j > 0; j >>= 1) {
      const int ixj = tid ^ j;
      if (ixj > tid) {
        const float ka = s_key[tid], kb = s_key[ixj];
        const int ia = s_idx[tid], ib = s_idx[ixj];
        const bool up = (tid & k) == 0;   // descending region
        const bool aG = (ka > kb) || (ka == kb && ia < ib);
        if (up ? !aG : aG) {
          s_key[tid] = kb; s_key[ixj] = ka;
          s_idx[tid] = ib; s_idx[ixj] = ia;
        }
      }
      __syncthreads();
    }
  }

  // 4) ordered probabilities
  const int sidx = s_idx[tid];
  const float pj = expf(s_lp[sidx]);
  s_rs[tid] = pj;

  // 5) permute first K (wave-0 bitonic on uniforms, ascending, via shfl_xor)
  if (tid < K_TOP) {
    const uint64_t np = (uint64_t)bid * K_TOP + (uint64_t)tid;
    float u = u01_from_bits(jax_bits(kp0, kp1, np, PERM_N2));
    float f = pj;
    for (int k = 2; k <= 32; k <<= 1) {
      for (int j = k >> 1; j > 0; j >>= 1) {
        const float u2 = __shfl_xor(u, j, 32);
        const float f2 = __shfl_xor(f, j, 32);
        const bool lower = (tid & j) == 0;
        const bool upDir = (tid & k) == 0;
        const bool takeMin = (upDir == lower);
        const bool take = takeMin ? (u2 < u) : (u2 > u);
        if (take) { u = u2; f = f2; }
      }
    }
    s_rs[tid] = f;
  }
  __syncthreads();

  // 6) suffix sums (denominators): Hillis-Steele over reversed order
  s_dn[tid] = s_rs[tid];
  __syncthreads();
  for (int d = 1; d < 512; d <<= 1) {
    float v = s_dn[tid];
    if (tid + d < 512) v += s_dn[tid + d];
    __syncthreads();
    s_dn[tid] = v;
    __syncthreads();
  }

  // 7) delta = sum(rel_g[:K]) - sum(rel_g[K:])  (reuse s_lp as reduce buffer)
  const float relg = s_rel[sidx];
  s_lp[tid] = (tid < K_TOP) ? relg : -relg;
  __syncthreads();
  for (int st = 256; st > 0; st >>= 1) {
    if (tid < st) s_lp[tid] += s_lp[tid + st];
    __syncthreads();
  }

  // 8) ranking prob as an f32 product (matches reference underflow behavior)
  if (tid < K_TOP) {
    float ratio = s_rs[tid] / s_dn[tid];
    for (int m = 16; m > 0; m >>= 1) ratio *= __shfl_xor(ratio, m, 32);
    if (tid == 0) {
      const float FACT_K = 2.6313083693369353e35f;  // 32!
      const float lprob = logf(FACT_K * ratio);
      gout[bid] = lprob * s_lp[0];
    }
  }
}

// ---- Kernel C: WMMA f32 row-sum reduction of the 102400 g values -----------
// D += A(16x4 chunk of g) x B(ones).  sum(D) == 16 * sum(A) regardless of the
// exact A/B lane layout, so this is layout-proof and stays in f32.
__global__ __launch_bounds__(32)
void reduce_wmma_kernel(const float* __restrict__ g, float* __restrict__ out) {
  const int lane = threadIdx.x;
  v8f acc0 = {0.f, 0.f, 0.f, 0.f, 0.f, 0.f, 0.f, 0.f};
  v8f acc1 = {0.f, 0.f, 0.f, 0.f, 0.f, 0.f, 0.f, 0.f};
  v2f ones; ones.x = 1.0f; ones.y = 1.0f;
  // A-matrix f32 16x4: lanes 0-15 hold K=0,1 for M=lane; lanes 16-31 K=2,3.
  const int off = (lane & 15) * 4 + ((lane >> 4) << 1);
  for (int c = 0; c < 1600; c += 2) {
    const float* p0 = g + (c + 0) * 64 + off;
    const float* p1 = g + (c + 1) * 64 + off;
    v2f a0; a0.x = p0[0]; a0.y = p0[1];
    v2f a1; a1.x = p1[0]; a1.y = p1[1];
    acc0 = __builtin_amdgcn_wmma_f32_16x16x4_f32(false, a0, false, ones,
                                                 (short)0, acc0, false, false);
    acc1 = __builtin_amdgcn_wmma_f32_16x16x4_f32(false, a1, false, ones,
                                                 (short)0, acc1, false, false);
  }
  v8f acc = acc0 + acc1;
  float t = 0.0f;
#pragma unroll
  for (int i = 0; i < 8; ++i) t += acc[i];
  for (int m = 16; m > 0; m >>= 1) t += __shfl_xor(t, m, 32);
  if (lane == 0)
    out[0] = -t / (16.0f * (float)B_ROWS * (float)NUM_MC);
}

extern "C" void kernel_launch(void* const* d_in, const int* in_sizes, int n_in,
                              void* d_out, int out_size, void* d_ws, size_t ws_size,
                              hipStream_t stream) {
  (void)in_sizes; (void)n_in; (void)out_size; (void)ws_size;
  const float* score = (const float*)d_in[0];
  const float* rel   = (const float*)d_in[1];
  float* out = (float*)d_out;
  float* ws  = (float*)d_ws;
  float* rowLse    = ws;            // 4096 floats
  float* rowRelSum = ws + B_ROWS;   // 4096 floats
  float* gbuf      = ws + 2 * B_ROWS; // 102400 floats

  // Derive kg/kp exactly like jax.random.split(jax.random.key(12345)):
  // counts iota(4) split into pairs (0,2) and (1,3).
  const uint32_t rk0 = 0u, rk1 = 12345u;
  U2 e0 = threefry2x32(rk0, rk1, 0u, 2u);
  U2 e1 = threefry2x32(rk0, rk1, 1u, 3u);
  const uint32_t kg0 = e0.a, kg1 = e1.a;   // bits[0], bits[1]
  const uint32_t kp0 = e0.b, kp1 = e1.b;   // bits[2], bits[3]

  row_stats_kernel<<<B_ROWS, 512, 0, stream>>>(score, rel, rowLse, rowRelSum);
  mc_task_kernel<<<NTASK, 512, 0, stream>>>(score, rel, rowLse, rowRelSum,
                                            kg0, kg1, kp0, kp1, gbuf);
  reduce_wmma_kernel<<<1, 32, 0, stream>>>(gbuf, out);
}